// SeparableRandomProjectLinearOperator_29171417874659
// MI455X (gfx1250) — compile-verified
//
#include <hip/hip_runtime.h>
#include <math.h>

typedef __attribute__((ext_vector_type(2))) float v2f;
typedef __attribute__((ext_vector_type(8))) float v8f;

#define KC    64   // HW chunk per LDS stage
#define PITCH 68   // LDS row pitch (floats), even (for 8B-aligned v2f) + bank spread

__device__ __forceinline__ float gelu_exact(float x) {
    // exact GELU (matches torch / jax approximate=False)
    return 0.5f * x * (1.0f + erff(x * 0.70710678118654752440f));
}

// One block per output pixel p = h*32+w. 256 threads = 8 wave32.
// G(64x64) = Co(64x1024) x V(1024x64) via V_WMMA_F32_16X16X4_F32,
// Co generated on the fly from the coordinate MLP (separable layer 1).
__global__ __launch_bounds__(256)
void srplo_fused_kernel(const float* __restrict__ v,
                        const float* __restrict__ params,
                        const float* __restrict__ w1,
                        const float* __restrict__ b1,
                        const float* __restrict__ w2,
                        const float* __restrict__ b2,
                        const float* __restrict__ down_w,
                        const float* __restrict__ down_b,
                        const float* __restrict__ bias_p,
                        float* __restrict__ out)
{
    __shared__ float sCo[64 * PITCH]; // coeff chunk Co[n][k]; reused as G[n][c] at the end
    __shared__ float sVt[64 * PITCH]; // V chunk transposed: sVt[c][k], c = b*32+i
    __shared__ float sW2[64 * 8];
    __shared__ float sB2[64];
    __shared__ float sT[64];

    const int t = threadIdx.x;
    const int p = blockIdx.x;      // output pixel
    const int h = p >> 5;
    const int w = p & 31;

    for (int idx = t; idx < 64 * 8; idx += 256) sW2[idx] = w2[idx];
    if (t < 64) sB2[t] = b2[t];

    // Separable first MLP layer: per-block (output pixel) partial pre-activation.
    const float ch = 2.0f * ((h + 0.5f) * (1.0f / 32.0f)) - 1.0f;
    const float cw = 2.0f * ((w + 0.5f) * (1.0f / 32.0f)) - 1.0f;
    float a_reg[8], w1H[8], w1W[8];
    #pragma unroll
    for (int j = 0; j < 8; ++j) {
        a_reg[j] = w1[j*4+0]*ch + w1[j*4+1]*cw + b1[j];
        w1H[j]   = w1[j*4+2];
        w1W[j]   = w1[j*4+3];
    }

    // Wave -> two 16x16 tiles of G (shared A fragment).
    const int lane  = t & 31;
    const int wv    = t >> 5;          // 0..7
    const int tm    = wv >> 1;         // tile row 0..3  (n block)
    const int tn0   = (wv & 1) * 2;    // tile cols tn0, tn0+1 (c block)
    const int half  = lane >> 4;       // 0: K pair {0,1}; 1: K pair {2,3}
    const int ln    = lane & 15;
    const int rowA  = tm * 16 + ln;
    const int colB0 = tn0 * 16 + ln;
    const int colB1 = colB0 + 16;

    v8f acc0 = {};
    v8f acc1 = {};

    const int   kpt  = t & 63;         // MLP point within chunk
    const int   ngrp = t >> 6;         // 4 groups x 16 n
    const int   cRow = t >> 2;         // V copy: column c = 0..63
    const int   cQ   = t & 3;          // V copy: 16-float segment
    const float inv  = 1.0f / 1024.0f; // fold the 1/(H*W) into V

    for (int c0 = 0; c0 < 1024; c0 += KC) {
        __syncthreads();               // LDS safe to overwrite

        // --- Phase 1: Co[n][k] for this HW chunk (MLP, layer1 separable) ---
        {
            const int   hw = c0 + kpt;
            const int   H  = hw >> 5, W = hw & 31;
            const float cH = 2.0f*((H + 0.5f)*(1.0f/32.0f)) - 1.0f;
            const float cW = 2.0f*((W + 0.5f)*(1.0f/32.0f)) - 1.0f;
            float hid[8];
            #pragma unroll
            for (int j = 0; j < 8; ++j)
                hid[j] = gelu_exact(a_reg[j] + w1H[j]*cH + w1W[j]*cW);
            #pragma unroll
            for (int nn = 0; nn < 16; ++nn) {
                const int n = ngrp*16 + nn;
                float s = sB2[n];
                #pragma unroll
                for (int j = 0; j < 8; ++j) s += sW2[n*8+j]*hid[j];
                sCo[n*PITCH + kpt] = s;
            }
        }

        // --- Phase 2: stage V chunk transposed (v layout IS [c][HW] already) ---
        {
            const float* src = v + cRow*1024 + c0 + cQ*16;
            float*       dst = &sVt[cRow*PITCH + cQ*16];
            #pragma unroll
            for (int u = 0; u < 4; ++u) {
                float4 x = *(const float4*)(src + u*4);
                dst[u*4+0] = x.x*inv; dst[u*4+1] = x.y*inv;
                dst[u*4+2] = x.z*inv; dst[u*4+3] = x.w*inv;
            }
            if (c0 + KC < 1024)
                __builtin_prefetch(src + KC, 0, 1);   // global_prefetch_b8 next chunk
        }
        __syncthreads();

        // --- Phase 3: WMMA accumulate (f32 16x16x4) ---
        const float* aBase  = &sCo[rowA  * PITCH];
        const float* b0Base = &sVt[colB0 * PITCH];
        const float* b1Base = &sVt[colB1 * PITCH];
        #pragma unroll
        for (int k0 = 0; k0 < KC; k0 += 4) {
            const int kk = k0 + 2*half;
            v2f a   = *(const v2f*)(aBase  + kk);
            v2f bb0 = *(const v2f*)(b0Base + kk);
            v2f bb1 = *(const v2f*)(b1Base + kk);
            acc0 = __builtin_amdgcn_wmma_f32_16x16x4_f32(
                       false, a, false, bb0, (short)0, acc0, false, false);
            acc1 = __builtin_amdgcn_wmma_f32_16x16x4_f32(
                       false, a, false, bb1, (short)0, acc1, false, false);
        }
    }

    __syncthreads();
    // Spill G tiles to LDS (reuse sCo). C/D layout: VGPR r -> M = r + 8*half.
    #pragma unroll
    for (int r = 0; r < 8; ++r) {
        const int row = tm*16 + r + 8*half;
        sCo[row*PITCH + colB0] = acc0[r];
        sCo[row*PITCH + colB1] = acc1[r];
    }
    __syncthreads();

    // t[b,i] = sum_n params[n,i] * G[n, b*32+i]
    if (t < 64) {
        const int i = t & 31;
        float s = 0.0f;
        #pragma unroll 8
        for (int n = 0; n < 64; ++n) s += params[n*32 + i] * sCo[n*PITCH + t];
        sT[t] = s;
    }
    __syncthreads();

    // out[b,o] = down_w . t[b,:] + down_b[o] + bias_p[o]
    if (t < 64) {
        const int b = t >> 5, o = t & 31;
        float s = down_b[o] + bias_p[o];
        #pragma unroll 8
        for (int i = 0; i < 32; ++i) s += down_w[o*32+i] * sT[b*32+i];
        out[(b*32 + o)*1024 + p] = s;
    }
}

extern "C" void kernel_launch(void* const* d_in, const int* in_sizes, int n_in,
                              void* d_out, int out_size, void* d_ws, size_t ws_size,
                              hipStream_t stream) {
    (void)in_sizes; (void)n_in; (void)out_size; (void)d_ws; (void)ws_size;
    const float* v      = (const float*)d_in[0];
    const float* params = (const float*)d_in[1];
    const float* w1     = (const float*)d_in[2];
    const float* b1     = (const float*)d_in[3];
    const float* w2     = (const float*)d_in[4];
    const float* b2     = (const float*)d_in[5];
    const float* down_w = (const float*)d_in[6];
    const float* down_b = (const float*)d_in[7];
    const float* bias_p = (const float*)d_in[8];
    float* out = (float*)d_out;

    srplo_fused_kernel<<<1024, 256, 0, stream>>>(
        v, params, w1, b1, w2, b2, down_w, down_b, bias_p, out);
}